// MultiheadSelfAttention_10900626997748
// MI455X (gfx1250) — compile-verified
//
#include <hip/hip_runtime.h>
#include <math.h>

// Problem constants (match reference)
#define BATCH   2
#define SEQ     2048
#define DMODEL  1024
#define NHEADS  16
#define DK      64
#define LOG_THETA 9.210340371976184f  // ln(10000)

typedef __attribute__((ext_vector_type(2))) float v2f;
typedef __attribute__((ext_vector_type(8))) float v8f;

static __device__ __forceinline__ v8f wmma_f32_4(v2f a, v2f b, v8f c) {
  // D = A(16x4) * B(4x16) + C(16x16), fp32 matrix core
  return __builtin_amdgcn_wmma_f32_16x16x4_f32(
      /*neg_a=*/false, a, /*neg_b=*/false, b,
      /*c_mod=*/(short)0, c, /*reuse_a=*/false, /*reuse_b=*/false);
}

// ---------------------------------------------------------------------------
// GEMM: Y[m,e] = sum_k A[m,k] * W[e,k]   (i.e. Y = A @ W^T), M=4096, N=K=1024
// One 16x16 output tile per wave; K-loop in steps of 4 on V_WMMA_F32_16X16X4.
// in_head_major:  A read as [B,H,S,DK] (attention output) instead of [m,1024]
// out_head_major: Y written as [B,H,S,DK] (for Q/K/V) instead of [m,1024]
// apply_rope:     fuse RoPE into the epilogue (for Q and K)
// ---------------------------------------------------------------------------
__global__ __launch_bounds__(256) void proj_gemm_kernel(
    const float* __restrict__ A, const float* __restrict__ W,
    float* __restrict__ Y, int in_head_major, int out_head_major,
    int apply_rope) {
  const int lane = threadIdx.x & 31;
  const int wave = threadIdx.x >> 5;
  const int l15  = lane & 15;
  const int half = lane >> 4;      // 0: lanes 0-15, 1: lanes 16-31
  const int koff = half * 2;       // A/B per-lane K offset per ISA layout
  const int n0 = (blockIdx.x * 8 + wave) * 16;   // output column tile
  const int m0 = blockIdx.y * 16;                // output row tile
  const int arow = m0 + l15;                     // this lane's A row (M)
  const int wrow = n0 + l15;                     // this lane's W row (N)

  v8f c = {0.f,0.f,0.f,0.f,0.f,0.f,0.f,0.f};

  if (in_head_major) {
    const int bb = arow >> 11;          // SEQ = 2048
    const int ss = arow & (SEQ - 1);
    #pragma unroll 8
    for (int kb = 0; kb < DMODEL; kb += 4) {
      const int k0 = kb + koff;         // even, k0%64 <= 62 -> pair in-head
      v2f a = *(const v2f*)(A + (((size_t)(bb * NHEADS + (k0 >> 6)) * SEQ + ss) * DK)
                              + (k0 & 63));
      v2f b = *(const v2f*)(W + (size_t)wrow * DMODEL + k0);
      c = wmma_f32_4(a, b, c);
    }
  } else {
    #pragma unroll 8
    for (int kb = 0; kb < DMODEL; kb += 4) {
      const int k0 = kb + koff;
      v2f a = *(const v2f*)(A + (size_t)arow * DMODEL + k0);
      v2f b = *(const v2f*)(W + (size_t)wrow * DMODEL + k0);
      c = wmma_f32_4(a, b, c);
    }
  }

  // Epilogue: C layout -> VGPR v holds (row = v + half*8, col = l15)
  const int e = n0 + l15;
  #pragma unroll
  for (int v = 0; v < 8; ++v) {
    const int mrow = m0 + v + half * 8;
    const int ss = mrow & (SEQ - 1);
    const int bb = mrow >> 11;
    float val = c[v];
    if (apply_rope) {
      // Pair partner column e^1 lives in lane^1 (16-aligned n-tile).
      float partner = __shfl_xor(val, 1, 32);
      const int p = (e & 63) >> 1;
      const float freq  = __expf(-(2.0f * LOG_THETA / (float)DK) * (float)p);
      const float phase = (float)ss * freq;
      float sn, cs;
      __sincosf(phase, &sn, &cs);
      // even: x_e*cos - x_o*sin ; odd: x_e*sin + x_o*cos
      val = (e & 1) ? (partner * sn + val * cs) : (val * cs - partner * sn);
    }
    if (out_head_major)
      Y[((size_t)(bb * NHEADS + (e >> 6)) * SEQ + ss) * DK + (e & 63)] = val;
    else
      Y[(size_t)mrow * DMODEL + e] = val;
  }
}

// ---------------------------------------------------------------------------
// Causal flash attention, fp32 WMMA. One wave per 16-query tile per (b,h).
// Q/K/V/O are [B, H, S, DK] head-major. Streaming online softmax; the 16x16
// score tile is staged through a per-wave LDS buffer for the cross-lane
// row-max / exp / row-sum phase, then fed back as the A operand of P*V.
// ---------------------------------------------------------------------------
__global__ __launch_bounds__(256) void attn_kernel(
    const float* __restrict__ Q, const float* __restrict__ K,
    const float* __restrict__ V, float* __restrict__ O) {
  __shared__ float ldsS[8][256];     // per-wave 16x16 score/P tile
  __shared__ float ldsFac[8][16];    // per-wave row rescale factors
  __shared__ float ldsStat[8][16];   // per-wave 1/l row normalizers

  const int lane = threadIdx.x & 31;
  const int wave = threadIdx.x >> 5;
  const int l15  = lane & 15;
  const int half = lane >> 4;
  const int koff = half * 2;
  const int qt = blockIdx.x * 8 + wave;          // query tile 0..127
  const int h  = blockIdx.y;
  const int b  = blockIdx.z;
  const size_t base = ((size_t)(b * NHEADS + h)) * SEQ * DK;
  const float* Qh = Q + base;
  const float* Kh = K + base;
  const float* Vh = V + base;
  float*       Oh = O + base;
  const int r0 = qt * 16;

  float* S    = ldsS[wave];
  float* FAC  = ldsFac[wave];
  float* STAT = ldsStat[wave];

  // Preload the 16x64 Q tile in WMMA A layout: 16 k-steps of float2 per lane.
  v2f qa[16];
  #pragma unroll
  for (int kb = 0; kb < 16; ++kb)
    qa[kb] = *(const v2f*)(Qh + (size_t)(r0 + l15) * DK + kb * 4 + koff);

  v8f o0 = {0.f,0.f,0.f,0.f,0.f,0.f,0.f,0.f};
  v8f o1 = o0, o2 = o0, o3 = o0;                 // 16x64 accumulator
  float m_i = -__builtin_inff();                 // lanes 0..15: row = lane
  float l_i = 0.f;

  for (int j = 0; j <= qt; ++j) {
    const int c0 = j * 16;

    // ---- S = Q * K^T (16x16), K-loop over 64 dims ----
    v8f s = {0.f,0.f,0.f,0.f,0.f,0.f,0.f,0.f};
    #pragma unroll
    for (int kb = 0; kb < 16; ++kb) {
      v2f kv = *(const v2f*)(Kh + (size_t)(c0 + l15) * DK + kb * 4 + koff);
      s = wmma_f32_4(qa[kb], kv, s);
    }

    // ---- scale + causal mask, stage to LDS ----
    const int kcol = c0 + l15;
    #pragma unroll
    for (int v = 0; v < 8; ++v) {
      const int qrow = r0 + v + half * 8;
      S[(v + half * 8) * 16 + l15] =
          (kcol <= qrow) ? s[v] * 0.125f : -__builtin_inff();
    }
    asm volatile("s_wait_dscnt 0" ::: "memory");

    // ---- online softmax row stats (one row per lane 0..15) ----
    if (lane < 16) {
      float mrow = -__builtin_inff();
      for (int c2 = 0; c2 < 16; ++c2) mrow = fmaxf(mrow, S[lane * 16 + c2]);
      const float mnew = fmaxf(m_i, mrow);
      const float fac  = __expf(m_i - mnew);     // exp(-inf)=0 on first tile
      float rs = 0.f;
      for (int c2 = 0; c2 < 16; ++c2) {
        const float p = __expf(S[lane * 16 + c2] - mnew);
        S[lane * 16 + c2] = p;                   // P overwrites S in place
        rs += p;
      }
      l_i = l_i * fac + rs;
      m_i = mnew;
      FAC[lane] = fac;
    }
    asm volatile("s_wait_dscnt 0" ::: "memory");

    // ---- rescale accumulators ----
    float facv[8];
    #pragma unroll
    for (int v = 0; v < 8; ++v) facv[v] = FAC[v + half * 8];
    #pragma unroll
    for (int v = 0; v < 8; ++v) {
      o0[v] *= facv[v]; o1[v] *= facv[v];
      o2[v] *= facv[v]; o3[v] *= facv[v];
    }

    // ---- O += P(16x16) * V(16x64): 4 k-steps x 4 n-groups of WMMA ----
    #pragma unroll
    for (int kb = 0; kb < 4; ++kb) {
      const int kk = kb * 4 + koff;
      v2f pa = *(const v2f*)(S + l15 * 16 + kk);            // A from LDS
      const float* vp = Vh + (size_t)(c0 + kk) * DK + l15;  // B from global
      v2f vb0 = { vp[0],  vp[DK] };
      v2f vb1 = { vp[16], vp[16 + DK] };
      v2f vb2 = { vp[32], vp[32 + DK] };
      v2f vb3 = { vp[48], vp[48 + DK] };
      o0 = wmma_f32_4(pa, vb0, o0);
      o1 = wmma_f32_4(pa, vb1, o1);
      o2 = wmma_f32_4(pa, vb2, o2);
      o3 = wmma_f32_4(pa, vb3, o3);
    }
  }

  // ---- normalize and write 16x64 output tile ----
  if (lane < 16) STAT[lane] = 1.0f / l_i;
  asm volatile("s_wait_dscnt 0" ::: "memory");
  #pragma unroll
  for (int v = 0; v < 8; ++v) {
    const float linv = STAT[v + half * 8];
    float* op = Oh + (size_t)(r0 + v + half * 8) * DK + l15;
    op[0]  = o0[v] * linv;
    op[16] = o1[v] * linv;
    op[32] = o2[v] * linv;
    op[48] = o3[v] * linv;
  }
}

// ---------------------------------------------------------------------------
extern "C" void kernel_launch(void* const* d_in, const int* in_sizes, int n_in,
                              void* d_out, int out_size, void* d_ws,
                              size_t ws_size, hipStream_t stream) {
  (void)in_sizes; (void)n_in; (void)out_size; (void)ws_size;
  const float* x  = (const float*)d_in[0];
  const float* Wq = (const float*)d_in[1];
  const float* Wk = (const float*)d_in[2];
  const float* Wv = (const float*)d_in[3];
  const float* Wo = (const float*)d_in[4];
  float* out = (float*)d_out;

  // Workspace: Q, K, V, attn-out, each [B,H,S,DK] fp32 = 16 MB (64 MB total).
  const size_t perBuf = (size_t)BATCH * NHEADS * SEQ * DK;
  float* Qb = (float*)d_ws;
  float* Kb = Qb + perBuf;
  float* Vb = Kb + perBuf;
  float* AO = Vb + perBuf;

  dim3 blk(256);
  dim3 gproj(8, (BATCH * SEQ) / 16, 1);  // 64 n-tiles (8 waves/blk) x 256 m-tiles

  // QKV projections (RoPE fused into Q and K epilogues)
  proj_gemm_kernel<<<gproj, blk, 0, stream>>>(x, Wq, Qb, 0, 1, 1);
  proj_gemm_kernel<<<gproj, blk, 0, stream>>>(x, Wk, Kb, 0, 1, 1);
  proj_gemm_kernel<<<gproj, blk, 0, stream>>>(x, Wv, Vb, 0, 1, 0);

  // Causal flash attention: 4096 independent waves (128 q-tiles x 16 h x 2 b)
  attn_kernel<<<dim3(SEQ / 16 / 8, NHEADS, BATCH), blk, 0, stream>>>(Qb, Kb, Vb, AO);

  // Output projection (reads head-major attention output)
  proj_gemm_kernel<<<gproj, blk, 0, stream>>>(AO, Wo, out, 1, 0, 0);
}